// TokenSwapMamba_5918464934174
// MI455X (gfx1250) — compile-verified
//
#include <hip/hip_runtime.h>
#include <hip/hip_bf16.h>
#include <math.h>

// ---------------------------------------------------------------------------
// TokenSwapMamba for MI455X (gfx1250, wave32, WMMA).
// B=2, L=8192, DIM=64, D_INNER=128, D_STATE=16, D_CONV=4, DT_RANK=4.
// All GEMMs use v_wmma_f32_16x16x4_f32 (exact fp32 => matches reference math).
// Selective scan is parallelized as a 3-pass chunked linear-recurrence scan.
// ---------------------------------------------------------------------------

#define BATCH   2
#define SEQ     8192
#define TOK     (BATCH * SEQ)          // 16384 tokens
#define DMODEL  64
#define DIN     128                    // D_INNER
#define DSTATE  16
#define DTRANK  4
#define XPROJ_N 36                     // DT_RANK + 2*D_STATE
#define XPROJ_P 48                     // padded to 3 WMMA tiles
#define CHUNK   128
#define NCH     (SEQ / CHUNK)          // 64 chunks

typedef __attribute__((ext_vector_type(2))) float v2f;
typedef __attribute__((ext_vector_type(8))) float v8f;

__device__ __forceinline__ v8f wmma4(v2f a, v2f b, v8f c) {
  // D = A(16x4) * B(4x16) + C(16x16), pure fp32 (CDNA5 V_WMMA_F32_16X16X4_F32)
  return __builtin_amdgcn_wmma_f32_16x16x4_f32(
      /*neg_a=*/false, a, /*neg_b=*/false, b,
      /*c_mod=*/(short)0, c, /*reuse_a=*/false, /*reuse_b=*/false);
}

__device__ __forceinline__ float sigmoidf_(float x) {
  return 1.0f / (1.0f + __expf(-x));
}
__device__ __forceinline__ float siluf_(float x) { return x * sigmoidf_(x); }
__device__ __forceinline__ float softplusf_(float x) {
  return (x > 20.0f) ? x : log1pf(__expf(x));
}

// ---------------------------------------------------------------------------
// Kernel 1: residual add + LayerNorm (64 feat) + half-channel swap.
// One wave (32 lanes) per token; each lane owns features (lane, lane+32).
// Writes the two residual outputs directly into d_out sections 2 & 3.
// ---------------------------------------------------------------------------
__global__ void prep_kernel(const float* __restrict__ under,
                            const float* __restrict__ over,
                            const float* __restrict__ ures_in,
                            const float* __restrict__ ores_in,
                            const float* __restrict__ n1w, const float* __restrict__ n1b,
                            const float* __restrict__ n2w, const float* __restrict__ n2b,
                            float* __restrict__ ures_out, float* __restrict__ ores_out,
                            float* __restrict__ xs_u, float* __restrict__ xs_o) {
  const int wave = threadIdx.x >> 5;
  const int lane = threadIdx.x & 31;
  const int tok  = blockIdx.x * 8 + wave;          // TOK divisible by 8
  const size_t base = (size_t)tok * DMODEL;
  const int f0 = lane, f1 = lane + 32;

  float u0 = under[base + f0] + ures_in[base + f0];
  float u1 = under[base + f1] + ures_in[base + f1];
  float o0 = over [base + f0] + ores_in[base + f0];
  float o1 = over [base + f1] + ores_in[base + f1];
  ures_out[base + f0] = u0; ures_out[base + f1] = u1;
  ores_out[base + f0] = o0; ores_out[base + f1] = o1;

  // mean over 64 features (wave32 butterfly)
  float su = u0 + u1, so = o0 + o1;
  #pragma unroll
  for (int off = 16; off > 0; off >>= 1) {
    su += __shfl_xor(su, off, 32);
    so += __shfl_xor(so, off, 32);
  }
  const float mu_u = su * (1.0f / 64.0f), mu_o = so * (1.0f / 64.0f);
  float vu = (u0 - mu_u) * (u0 - mu_u) + (u1 - mu_u) * (u1 - mu_u);
  float vo = (o0 - mu_o) * (o0 - mu_o) + (o1 - mu_o) * (o1 - mu_o);
  #pragma unroll
  for (int off = 16; off > 0; off >>= 1) {
    vu += __shfl_xor(vu, off, 32);
    vo += __shfl_xor(vo, off, 32);
  }
  const float iu = rsqrtf(vu * (1.0f / 64.0f) + 1e-5f);
  const float io = rsqrtf(vo * (1.0f / 64.0f) + 1e-5f);

  const float un0 = (u0 - mu_u) * iu * n1w[f0] + n1b[f0];
  const float un1 = (u1 - mu_u) * iu * n1w[f1] + n1b[f1];
  const float ov0 = (o0 - mu_o) * io * n2w[f0] + n2b[f0];
  const float ov1 = (o1 - mu_o) * io * n2w[f1] + n2b[f1];

  // swap: under_swap = [ov[:,:32], un[:,32:]], over_swap = [un[:,:32], ov[:,32:]]
  xs_u[base + f0] = ov0;  xs_u[base + f1] = un1;
  xs_o[base + f0] = un0;  xs_o[base + f1] = ov1;
}

// ---------------------------------------------------------------------------
// Kernel 2: in-proj GEMM  xz[TOK,256] = X[TOK,64] * in_w[256,64]^T   (WMMA)
// 256 threads = 8 waves; each wave: one 16-token M tile x two 16-col N tiles.
// ---------------------------------------------------------------------------
__global__ void gemm_in(const float* __restrict__ X, const float* __restrict__ W,
                        float* __restrict__ Y) {
  const int lane = threadIdx.x & 31;
  const int wave = threadIdx.x >> 5;       // 0..7
  const int t0   = blockIdx.x * 16;
  const int hi   = lane >> 4;              // half-wave
  const int lr   = lane & 15;
  const float* xrow = X + (size_t)(t0 + lr) * 64;

  v8f acc[2];
  #pragma unroll
  for (int nt = 0; nt < 2; ++nt) {
    const int e0 = (wave * 2 + nt) * 16;
    const float* wrow = W + (size_t)(e0 + lr) * 64;
    v8f c = {};
    #pragma unroll
    for (int k0 = 0; k0 < 64; k0 += 4) {
      v2f a, b;
      a.x = xrow[k0 + 2 * hi + 0]; a.y = xrow[k0 + 2 * hi + 1];
      b.x = wrow[k0 + 2 * hi + 0]; b.y = wrow[k0 + 2 * hi + 1];
      c = wmma4(a, b, c);
    }
    acc[nt] = c;
  }
  #pragma unroll
  for (int nt = 0; nt < 2; ++nt) {
    const int e0 = (wave * 2 + nt) * 16;
    #pragma unroll
    for (int r = 0; r < 8; ++r)
      Y[(size_t)(t0 + r + 8 * hi) * 256 + e0 + lr] = acc[nt][r];
  }
}

// ---------------------------------------------------------------------------
// Kernel 3: depthwise causal conv (k=4) + bias + SiLU.  xi = xz[:, :128]
// ---------------------------------------------------------------------------
__global__ void conv_silu(const float* __restrict__ xz, const float* __restrict__ cw,
                          const float* __restrict__ cb, float* __restrict__ xa) {
  const int gid = blockIdx.x * 256 + threadIdx.x;   // over TOK*128
  const int d = gid & 127;
  const int t = gid >> 7;
  const int l = t & (SEQ - 1);
  const int b = t >> 13;
  float acc = cb[d];
  #pragma unroll
  for (int j = 0; j < 4; ++j) {
    const int ll = l - 3 + j;
    if (ll >= 0)
      acc += cw[d * 4 + j] * xz[((size_t)(b * SEQ + ll)) * 256 + d];
  }
  xa[(size_t)t * DIN + d] = siluf_(acc);
}

// ---------------------------------------------------------------------------
// Kernel 4: x-proj GEMM  xdbl[TOK,48] = xa[TOK,128] * xproj_w[36,128]^T (WMMA)
// N padded 36->48 (3 tiles); 96 threads = 3 waves, one N tile per wave.
// ---------------------------------------------------------------------------
__global__ void gemm_xproj(const float* __restrict__ X, const float* __restrict__ W,
                           float* __restrict__ Y) {
  const int lane = threadIdx.x & 31;
  const int wave = threadIdx.x >> 5;       // 0..2
  const int t0   = blockIdx.x * 16;
  const int hi   = lane >> 4;
  const int lr   = lane & 15;
  const int e0   = wave * 16;
  const int er   = e0 + lr;
  const bool ok  = (er < XPROJ_N);
  const float* xrow = X + (size_t)(t0 + lr) * DIN;
  const float* wrow = W + (size_t)(ok ? er : 0) * DIN;
  v8f c = {};
  #pragma unroll
  for (int k0 = 0; k0 < DIN; k0 += 4) {
    v2f a, b;
    a.x = xrow[k0 + 2 * hi + 0]; a.y = xrow[k0 + 2 * hi + 1];
    b.x = ok ? wrow[k0 + 2 * hi + 0] : 0.0f;
    b.y = ok ? wrow[k0 + 2 * hi + 1] : 0.0f;
    c = wmma4(a, b, c);
  }
  #pragma unroll
  for (int r = 0; r < 8; ++r)
    Y[(size_t)(t0 + r + 8 * hi) * XPROJ_P + e0 + lr] = c[r];
}

// ---------------------------------------------------------------------------
// Selective scan, pass A: per-chunk local scan + decay product.
// Block = 128 threads (one per channel d), grid = (NCH, BATCH).
// Each thread keeps all 16 states in registers; xdbl rows staged in LDS.
//   P[s] = prod_t exp(delta_t * A[d,s]);  Q[s] = local scan ending state (h_in=0)
// ---------------------------------------------------------------------------
__global__ void scan_passA(const float* __restrict__ xdbl, const float* __restrict__ xa,
                           const float* __restrict__ dtw, const float* __restrict__ dtb,
                           const float* __restrict__ A_log,
                           float* __restrict__ P, float* __restrict__ Q) {
  __shared__ float sx[CHUNK * XPROJ_P];            // 24 KB of 320 KB WGP LDS
  const int d = threadIdx.x;
  const int g = blockIdx.x, b = blockIdx.y;
  const int l0 = g * CHUNK;
  const float* src = xdbl + ((size_t)b * SEQ + l0) * XPROJ_P;
  for (int i = threadIdx.x; i < CHUNK * XPROJ_P; i += 128) sx[i] = src[i];
  __syncthreads();

  const float w0 = dtw[d * 4 + 0], w1 = dtw[d * 4 + 1];
  const float w2 = dtw[d * 4 + 2], w3 = dtw[d * 4 + 3];
  const float bb = dtb[d];
  float As[DSTATE], Ps[DSTATE], Qs[DSTATE];
  #pragma unroll
  for (int s = 0; s < DSTATE; ++s) {
    As[s] = -__expf(A_log[d * DSTATE + s]);
    Ps[s] = 1.0f; Qs[s] = 0.0f;
  }
  const float* xap = xa + ((size_t)b * SEQ + l0) * DIN + d;
  for (int t = 0; t < CHUNK; ++t) {
    const float* r = sx + t * XPROJ_P;
    const float delta = softplusf_(r[0] * w0 + r[1] * w1 + r[2] * w2 + r[3] * w3 + bb);
    const float du = delta * xap[(size_t)t * DIN];
    #pragma unroll
    for (int s = 0; s < DSTATE; ++s) {
      const float a = __expf(delta * As[s]);
      Qs[s] = a * Qs[s] + du * r[DTRANK + s];
      Ps[s] *= a;
    }
  }
  const size_t base = (((size_t)b * NCH + g) * DIN + d) * DSTATE;
  #pragma unroll
  for (int s = 0; s < DSTATE; ++s) { P[base + s] = Ps[s]; Q[base + s] = Qs[s]; }
}

// ---------------------------------------------------------------------------
// Pass B: serial composition over the 64 chunk carries (4096 parallel chains).
//   hin[g] = incoming state of chunk g;  h <- P[g]*h + Q[g]
// ---------------------------------------------------------------------------
__global__ void scan_passB(const float* __restrict__ P, const float* __restrict__ Q,
                           float* __restrict__ hin) {
  const int gt = blockIdx.x * 256 + threadIdx.x;   // 0..4095
  const int b  = gt >> 11;
  const int ds = gt & 2047;
  float h = 0.0f;
  for (int g = 0; g < NCH; ++g) {
    const size_t idx = ((size_t)b * NCH + g) * (DIN * DSTATE) + ds;
    hin[idx] = h;
    h = P[idx] * h + Q[idx];
  }
}

// ---------------------------------------------------------------------------
// Pass C: replay chunk with correct incoming state; fuse y = sum_s h*C,
// + xa*D, and gate by silu(z).  Produces y_gated[TOK,128].
// ---------------------------------------------------------------------------
__global__ void scan_passC(const float* __restrict__ xdbl, const float* __restrict__ xa,
                           const float* __restrict__ xz,
                           const float* __restrict__ dtw, const float* __restrict__ dtb,
                           const float* __restrict__ A_log, const float* __restrict__ Dv,
                           const float* __restrict__ hin, float* __restrict__ yg) {
  __shared__ float sx[CHUNK * XPROJ_P];
  const int d = threadIdx.x;
  const int g = blockIdx.x, b = blockIdx.y;
  const int l0 = g * CHUNK;
  const float* src = xdbl + ((size_t)b * SEQ + l0) * XPROJ_P;
  for (int i = threadIdx.x; i < CHUNK * XPROJ_P; i += 128) sx[i] = src[i];
  __syncthreads();

  const float w0 = dtw[d * 4 + 0], w1 = dtw[d * 4 + 1];
  const float w2 = dtw[d * 4 + 2], w3 = dtw[d * 4 + 3];
  const float bb = dtb[d];
  const float Dd = Dv[d];
  float As[DSTATE], h[DSTATE];
  const size_t hbase = (((size_t)b * NCH + g) * DIN + d) * DSTATE;
  #pragma unroll
  for (int s = 0; s < DSTATE; ++s) {
    As[s] = -__expf(A_log[d * DSTATE + s]);
    h[s]  = hin[hbase + s];
  }
  for (int t = 0; t < CHUNK; ++t) {
    const size_t tok = (size_t)b * SEQ + l0 + t;
    const float* r = sx + t * XPROJ_P;
    const float delta = softplusf_(r[0] * w0 + r[1] * w1 + r[2] * w2 + r[3] * w3 + bb);
    const float xad = xa[tok * DIN + d];
    const float du = delta * xad;
    float y = 0.0f;
    #pragma unroll
    for (int s = 0; s < DSTATE; ++s) {
      const float a = __expf(delta * As[s]);
      h[s] = a * h[s] + du * r[DTRANK + s];              // B_t = xdbl[4+s]
      y += h[s] * r[DTRANK + DSTATE + s];                // C_t = xdbl[20+s]
    }
    y += xad * Dd;
    const float z = xz[tok * 256 + DIN + d];
    yg[tok * DIN + d] = y * siluf_(z);
  }
}

// ---------------------------------------------------------------------------
// Kernel 8: out-proj GEMM  out[TOK,64] = yg[TOK,128] * out_w[64,128]^T (WMMA)
// 128 threads = 4 waves, one 16-col N tile each.
// ---------------------------------------------------------------------------
__global__ void gemm_out(const float* __restrict__ X, const float* __restrict__ W,
                         float* __restrict__ Y) {
  const int lane = threadIdx.x & 31;
  const int wave = threadIdx.x >> 5;       // 0..3
  const int t0   = blockIdx.x * 16;
  const int hi   = lane >> 4;
  const int lr   = lane & 15;
  const int e0   = wave * 16;
  const float* xrow = X + (size_t)(t0 + lr) * DIN;
  const float* wrow = W + (size_t)(e0 + lr) * DIN;
  v8f c = {};
  #pragma unroll
  for (int k0 = 0; k0 < DIN; k0 += 4) {
    v2f a, b;
    a.x = xrow[k0 + 2 * hi + 0]; a.y = xrow[k0 + 2 * hi + 1];
    b.x = wrow[k0 + 2 * hi + 0]; b.y = wrow[k0 + 2 * hi + 1];
    c = wmma4(a, b, c);
  }
  #pragma unroll
  for (int r = 0; r < 8; ++r)
    Y[(size_t)(t0 + r + 8 * hi) * DMODEL + e0 + lr] = c[r];
}

// ---------------------------------------------------------------------------
// Host-side launch
// ---------------------------------------------------------------------------
extern "C" void kernel_launch(void* const* d_in, const int* in_sizes, int n_in,
                              void* d_out, int out_size, void* d_ws, size_t ws_size,
                              hipStream_t stream) {
  (void)in_sizes; (void)n_in; (void)out_size; (void)ws_size;

  const float* under = (const float*)d_in[0];
  const float* over  = (const float*)d_in[1];
  const float* uresI = (const float*)d_in[2];
  const float* oresI = (const float*)d_in[3];
  const float* n1w = (const float*)d_in[4];
  const float* n1b = (const float*)d_in[5];
  const float* n2w = (const float*)d_in[6];
  const float* n2b = (const float*)d_in[7];

  float* out = (float*)d_out;
  const size_t SEC = (size_t)TOK * DMODEL;           // 1,048,576
  float* out_uswap = out;                            // under_swap
  float* out_oswap = out + SEC;                      // over_swap
  float* out_ures  = out + 2 * SEC;                  // under_residual
  float* out_ores  = out + 3 * SEC;                  // over_residual

  // workspace layout (floats). total ~12.06M floats (~46 MB)
  float* ws = (float*)d_ws;
  float* xs_u = ws;                                  // TOK*64
  float* xs_o = xs_u + SEC;                          // TOK*64
  float* xz   = xs_o + SEC;                          // TOK*256
  float* xa   = xz   + (size_t)TOK * 256;            // TOK*128
  float* xdbl = xa   + (size_t)TOK * DIN;            // TOK*48
  float* yg   = xdbl + (size_t)TOK * XPROJ_P;        // TOK*128
  float* Pc   = yg   + (size_t)TOK * DIN;            // B*NCH*128*16
  float* Qc   = Pc   + (size_t)BATCH * NCH * DIN * DSTATE;
  float* hin  = Qc   + (size_t)BATCH * NCH * DIN * DSTATE;

  // 1) residual add + LN + swap (also writes residual outputs)
  prep_kernel<<<TOK / 8, 256, 0, stream>>>(under, over, uresI, oresI,
                                           n1w, n1b, n2w, n2b,
                                           out_ures, out_ores, xs_u, xs_o);

  // 2) two Mamba streams, sequential, reusing the same scratch
  for (int sidx = 0; sidx < 2; ++sidx) {
    const int pb = 8 + sidx * 9;
    const float* in_w    = (const float*)d_in[pb + 0];
    const float* conv_w  = (const float*)d_in[pb + 1];
    const float* conv_b  = (const float*)d_in[pb + 2];
    const float* xproj_w = (const float*)d_in[pb + 3];
    const float* dt_w    = (const float*)d_in[pb + 4];
    const float* dt_b    = (const float*)d_in[pb + 5];
    const float* A_log   = (const float*)d_in[pb + 6];
    const float* Dv      = (const float*)d_in[pb + 7];
    const float* out_w   = (const float*)d_in[pb + 8];
    const float* xs      = sidx ? xs_o : xs_u;
    float*       outsec  = sidx ? out_oswap : out_uswap;

    gemm_in   <<<TOK / 16, 256, 0, stream>>>(xs, in_w, xz);
    conv_silu <<<(TOK * DIN) / 256, 256, 0, stream>>>(xz, conv_w, conv_b, xa);
    gemm_xproj<<<TOK / 16, 96, 0, stream>>>(xa, xproj_w, xdbl);
    scan_passA<<<dim3(NCH, BATCH), 128, 0, stream>>>(xdbl, xa, dt_w, dt_b, A_log, Pc, Qc);
    scan_passB<<<(BATCH * DIN * DSTATE) / 256, 256, 0, stream>>>(Pc, Qc, hin);
    scan_passC<<<dim3(NCH, BATCH), 128, 0, stream>>>(xdbl, xa, xz, dt_w, dt_b,
                                                     A_log, Dv, hin, yg);
    gemm_out  <<<TOK / 16, 128, 0, stream>>>(yg, out_w, outsec);
  }
}